// GatedGraphConvolution_76081050681489
// MI455X (gfx1250) — compile-verified
//
#include <hip/hip_runtime.h>

typedef __attribute__((ext_vector_type(2))) float v2f;
typedef __attribute__((ext_vector_type(8))) float v8f;

#define BB 64
#define NN 1024
#define CC 128
#define C3 384
#define KC 32      // K-chunk for the bmm (double-buffered)
#define SA 36      // LDS stride for adj tile: 36*r mod 64 distinct for r<16
#define SB 136     // LDS stride for ann tile: half-waves hit disjoint bank groups
#define SV 132     // LDS stride for v tile

static __device__ __forceinline__ v8f wmma_f32(v2f a, v2f b, v8f c) {
    return __builtin_amdgcn_wmma_f32_16x16x4_f32(false, a, false, b, (short)0, c,
                                                 false, false);
}

// CDNA5 async copy: global -> LDS, 16B per lane, tracked by ASYNCcnt.
static __device__ __forceinline__ void async_ld_b128(unsigned ldsOff, unsigned gOff,
                                                     const void* sbase) {
    asm volatile("global_load_async_to_lds_b128 %0, %1, %2 offset:0"
                 :: "v"(ldsOff), "v"(gOff), "s"(sbase)
                 : "memory");
}
static __device__ __forceinline__ void wait_async0() {
    asm volatile("s_wait_asynccnt 0x0" ::: "memory");
}
// low 32 bits of a generic shared pointer == LDS byte offset (aperture scheme)
static __device__ __forceinline__ unsigned lds_off(const void* p) {
    return (unsigned)(uintptr_t)p;
}

__launch_bounds__(256, 1)
__global__ void ggnn_fused_kernel(const float* __restrict__ adj,   // [B,N,N]
                                  const float* __restrict__ ann,   // [B,N,C]
                                  const float* __restrict__ gcb,   // [C]
                                  const float* __restrict__ Wk,    // [C,3C]
                                  const float* __restrict__ Wr,    // [C,3C]
                                  const float* __restrict__ gbias, // [2,3C]
                                  float* __restrict__ out)         // [B,N,C]
{
    __shared__ float adjLds[2][32 * SA];   // 32 rows x 32 K, double buffered
    __shared__ float annLds[2][KC * SB];   // 32 K x 128 C, double buffered
    __shared__ float vLds[32 * SV];        // 32 rows x 128 C

    const int b    = blockIdx.y;
    const int m0   = blockIdx.x << 5;          // row block of 32 graph nodes
    const int tid  = threadIdx.x;
    const int lane = tid & 31;
    const int w    = tid >> 5;                 // wave id 0..7 -> column tile
    const int r    = lane & 15;
    const int half = lane >> 4;                // 0: lanes 0-15, 1: lanes 16-31
    const int n0   = w << 4;

    const float* adjB = adj + (size_t)b * NN * NN + (size_t)m0 * NN;
    const float* annB = ann + (size_t)b * NN * CC;

    // thread's fixed slice of each staged chunk
    const int aRow = tid >> 3, aCol = (tid & 7) << 2;   // adj: 256 float4, 1/thread

    // ---------------- Phase 1: v = adj @ ann, two 16x16 tiles per wave
    v8f acc0, acc1;
#pragma unroll
    for (int i = 0; i < 8; ++i) { acc0[i] = 0.0f; acc1[i] = 0.0f; }

    // prologue: stage chunk 0 into buffer 0
    async_ld_b128(lds_off(&adjLds[0][aRow * SA + aCol]),
                  (unsigned)((aRow * NN + aCol) * 4), adjB);
#pragma unroll
    for (int i = 0; i < 4; ++i) {              // ann: 1024 float4, 4/thread
        int e = tid + (i << 8);
        int row = e >> 5, col = (e & 31) << 2;
        async_ld_b128(lds_off(&annLds[0][row * SB + col]),
                      (unsigned)((row * CC + col) * 4), annB);
    }
    wait_async0();
    __syncthreads();

    for (int kc = 0; kc < NN; kc += KC) {
        const int cur = (kc >> 5) & 1;
        // issue next chunk's async copies into the alternate buffer (overlap)
        if (kc + KC < NN) {
            const int nxt = cur ^ 1;
            async_ld_b128(lds_off(&adjLds[nxt][aRow * SA + aCol]),
                          (unsigned)((aRow * NN + kc + KC + aCol) * 4), adjB);
#pragma unroll
            for (int i = 0; i < 4; ++i) {
                int e = tid + (i << 8);
                int row = e >> 5, col = (e & 31) << 2;
                async_ld_b128(lds_off(&annLds[nxt][row * SB + col]),
                              (unsigned)(((kc + KC + row) * CC + col) * 4), annB);
            }
        }
        if (kc + 2 * KC < NN)                  // L2 prefetch two chunks ahead
            __builtin_prefetch(adjB + (size_t)aRow * NN + kc + 2 * KC + aCol, 0, 0);

        const float* adjC = adjLds[cur];
        const float* annC = annLds[cur];
#pragma unroll
        for (int k0 = 0; k0 < KC; k0 += 4) {
            const int ka = k0 + (half << 1);
            v2f a0, a1, bf;
            // A 16x4: lanes 0-15 -> K=k0,k0+1 ; lanes 16-31 -> K=k0+2,k0+3
            a0.x = adjC[r * SA + ka];          a0.y = adjC[r * SA + ka + 1];
            a1.x = adjC[(r + 16) * SA + ka];   a1.y = adjC[(r + 16) * SA + ka + 1];
            // B 4x16 shared by both row tiles
            bf.x = annC[ka * SB + n0 + r];
            bf.y = annC[(ka + 1) * SB + n0 + r];
            acc0 = wmma_f32(a0, bf, acc0);     // independent chains
            acc1 = wmma_f32(a1, bf, acc1);
        }
        wait_async0();     // usually already satisfied: overlapped with WMMAs
        __syncthreads();   // all waves done reading cur + next buffer complete
    }

    // + gc_bias (each lane's C/D regs all sit in column n0+r)
    const float bias = gcb[n0 + r];
#pragma unroll
    for (int i = 0; i < 8; ++i) { acc0[i] += bias; acc1[i] += bias; }

    // stage v (32 x 128) in LDS: A-matrix of the GRU GEMMs
#pragma unroll
    for (int i = 0; i < 8; ++i) {
        vLds[(i + (half << 3)) * SV + n0 + r]      = acc0[i];
        vLds[(16 + i + (half << 3)) * SV + n0 + r] = acc1[i];
    }
    __syncthreads();

    // ---------------- Phase 2: GRU gates. xz+hz / xr+hr fused in one C chain.
    const int cz = n0 + r;  // column inside a gate block
    v8f z0, z1, rg0, rg1, xh0, xh1, hi0, hi1;
    const float bz  = gbias[cz] + gbias[C3 + cz];
    const float brr = gbias[CC + cz] + gbias[C3 + CC + cz];
    const float b0h = gbias[2 * CC + cz];
    const float b1h = gbias[C3 + 2 * CC + cz];
#pragma unroll
    for (int i = 0; i < 8; ++i) {
        z0[i] = bz;  z1[i] = bz;  rg0[i] = brr; rg1[i] = brr;
        xh0[i] = b0h; xh1[i] = b0h; hi0[i] = b1h; hi1[i] = b1h;
    }

    const float* hrow = annB + (size_t)m0 * CC;   // h tile = annotations rows m0..m0+31

#pragma unroll 2
    for (int kk = 0; kk < CC; kk += 4) {
        const int ka = kk + (half << 1);
        v2f av0, av1, ah0, ah1;
        av0.x = vLds[r * SV + ka];         av0.y = vLds[r * SV + ka + 1];
        av1.x = vLds[(r + 16) * SV + ka];  av1.y = vLds[(r + 16) * SV + ka + 1];
        ah0.x = hrow[(size_t)r * CC + ka];         ah0.y = hrow[(size_t)r * CC + ka + 1];
        ah1.x = hrow[(size_t)(r + 16) * CC + ka];  ah1.y = hrow[(size_t)(r + 16) * CC + ka + 1];

        const float* k0p = Wk + (size_t)ka * C3;
        const float* k1p = Wk + (size_t)(ka + 1) * C3;
        const float* r0p = Wr + (size_t)ka * C3;
        const float* r1p = Wr + (size_t)(ka + 1) * C3;
        v2f bkz, bkr, bkh, brz, brrf, brh;
        bkz.x  = k0p[cz];            bkz.y  = k1p[cz];
        bkr.x  = k0p[CC + cz];       bkr.y  = k1p[CC + cz];
        bkh.x  = k0p[2 * CC + cz];   bkh.y  = k1p[2 * CC + cz];
        brz.x  = r0p[cz];            brz.y  = r1p[cz];
        brrf.x = r0p[CC + cz];       brrf.y = r1p[CC + cz];
        brh.x  = r0p[2 * CC + cz];   brh.y  = r1p[2 * CC + cz];

        z0  = wmma_f32(av0, bkz, z0);   z1  = wmma_f32(av1, bkz, z1);
        z0  = wmma_f32(ah0, brz, z0);   z1  = wmma_f32(ah1, brz, z1);
        rg0 = wmma_f32(av0, bkr, rg0);  rg1 = wmma_f32(av1, bkr, rg1);
        rg0 = wmma_f32(ah0, brrf, rg0); rg1 = wmma_f32(ah1, brrf, rg1);
        xh0 = wmma_f32(av0, bkh, xh0);  xh1 = wmma_f32(av1, bkh, xh1);
        hi0 = wmma_f32(ah0, brh, hi0);  hi1 = wmma_f32(ah1, brh, hi1);
    }

    // ---------------- Phase 3: gate math + store (layouts already aligned)
    float* outB = out + (size_t)b * NN * CC + (size_t)m0 * CC + n0 + r;
#pragma unroll
    for (int i = 0; i < 8; ++i) {
        {   // rows m0 + i + 8*half
            const int row = i + (half << 3);
            const float hv  = hrow[(size_t)row * CC + n0 + r];
            const float z   = 1.0f / (1.0f + __expf(-z0[i]));
            const float rr  = 1.0f / (1.0f + __expf(-rg0[i]));
            const float pre = xh0[i] + rr * hi0[i];
            const float e2  = __expf(-2.0f * pre);
            const float th  = (1.0f - e2) / (1.0f + e2);
            outB[(size_t)row * CC] = z * hv + (1.0f - z) * th;
        }
        {   // rows m0 + 16 + i + 8*half
            const int row = 16 + i + (half << 3);
            const float hv  = hrow[(size_t)row * CC + n0 + r];
            const float z   = 1.0f / (1.0f + __expf(-z1[i]));
            const float rr  = 1.0f / (1.0f + __expf(-rg1[i]));
            const float pre = xh1[i] + rr * hi1[i];
            const float e2  = __expf(-2.0f * pre);
            const float th  = (1.0f - e2) / (1.0f + e2);
            outB[(size_t)row * CC] = z * hv + (1.0f - z) * th;
        }
    }
}

extern "C" void kernel_launch(void* const* d_in, const int* in_sizes, int n_in,
                              void* d_out, int out_size, void* d_ws, size_t ws_size,
                              hipStream_t stream) {
    (void)in_sizes; (void)n_in; (void)out_size; (void)d_ws; (void)ws_size;
    const float* adj   = (const float*)d_in[0];
    const float* ann   = (const float*)d_in[1];
    const float* gcb   = (const float*)d_in[2];
    const float* Wk    = (const float*)d_in[3];
    const float* Wr    = (const float*)d_in[4];
    const float* gbias = (const float*)d_in[5];
    float* out = (float*)d_out;

    dim3 grid(NN / 32, BB);   // 32 row-blocks x 64 batches
    dim3 block(256);          // 8 wave32s, one 16-col tile (x2 row tiles) each
    ggnn_fused_kernel<<<grid, block, 0, stream>>>(adj, ann, gcb, Wk, Wr, gbias, out);
}